// get_loss_85152021610990
// MI455X (gfx1250) — compile-verified
//
#include <hip/hip_runtime.h>
#include <hip/hip_bf16.h>

typedef __attribute__((ext_vector_type(2))) float v2f;
typedef __attribute__((ext_vector_type(8))) float v8f;

// Generic one-directional chamfer: for each point i in A (NA points), find
// min_j ||A_i - B_j||^2 over B (NB points), then atomically accumulate
// (sum over i of min) * invCount into *acc.  Point (b, i, k) of a cloud is at
// base + b*sB + i*sI + k*sK, so both [B,3,N] (channel-major) and [B,M,3]
// (point-major) layouts are handled by strides.
//
// One wave = one 16-row tile of A; loops over NB in 16-col tiles.
// Cross terms via V_WMMA_F32_16X16X4_F32 (K padded 3 -> 4 with zeros).
__global__ __launch_bounds__(128)
void chamfer_rowmin_kernel(const float* __restrict__ A, int sAB, int sAI, int sAK, int NA,
                           const float* __restrict__ Bp, int sBB, int sBI, int sBK, int NB,
                           float* __restrict__ acc, float invCount)
{
    const int lane = threadIdx.x & 31;
    const int wave = threadIdx.x >> 5;
    const int b    = blockIdx.y;
    const int r0   = blockIdx.x * 64 + wave * 16;   // 4 waves/block * 16 rows

    const float* Ab = A  + b * sAB;
    const float* Bb = Bp + b * sBB;

    const int hi = lane >> 4;     // 0: K=0,1 half   1: K=2,3 half
    const int lo = lane & 15;

    // ---- A fragment: 16x4 f32, M = r0+lo, K striped across lane halves ----
    const float* ap = Ab + (r0 + lo) * sAI;
    float a0 = ap[0], a1 = ap[sAK], a2 = ap[2 * sAK];
    v2f afrag;
    afrag.x = hi ? a2  : a0;      // K=2 : K=0
    afrag.y = hi ? 0.f : a1;      // K=3(pad) : K=1

    // ---- squared norms of the 8 A-rows this lane-half covers (D layout) ----
    float rn[8];
#pragma unroll
    for (int v = 0; v < 8; ++v) {
        const float* rp = Ab + (r0 + v + 8 * hi) * sAI;
        float x = rp[0], y = rp[sAK], z = rp[2 * sAK];
        rn[v] = x * x + y * y + z * z;
    }

    float rowmin[8];
#pragma unroll
    for (int v = 0; v < 8; ++v) rowmin[v] = 3.0e38f;

    for (int c0 = 0; c0 < NB; c0 += 16) {
        // ---- B fragment: 4x16 f32, N = c0+lo, K striped across lane halves ----
        const float* bp = Bb + (c0 + lo) * sBI;
        float b0 = bp[0], b1 = bp[sBK], b2 = bp[2 * sBK];
        v2f bfrag;
        bfrag.x = hi ? b2  : b0;
        bfrag.y = hi ? 0.f : b1;
        float cn = b0 * b0 + b1 * b1 + b2 * b2;   // ||B_j||^2, j = c0+lo (col of this lane)

        v8f cz = {};
        // cross[v] = dot(A_row(v,lane), B_col(lane))   (exact f32, K=4 w/ zero pad)
        v8f cross = __builtin_amdgcn_wmma_f32_16x16x4_f32(
            false, afrag, false, bfrag, (short)0, cz, false, false);

#pragma unroll
        for (int v = 0; v < 8; ++v) {
            float d = fmaf(-2.0f, cross[v], rn[v] + cn);
            d = fmaxf(d, 0.0f);
            rowmin[v] = fminf(rowmin[v], d);
        }
    }

    // ---- reduce the column dimension: min across the 16 lanes of each half ----
#pragma unroll
    for (int m = 1; m < 16; m <<= 1) {
#pragma unroll
        for (int v = 0; v < 8; ++v)
            rowmin[v] = fminf(rowmin[v], __shfl_xor(rowmin[v], m, 32));
    }

    if (lo == 0) {   // lanes 0 and 16: rows r0..r0+7 and r0+8..r0+15
        float s = 0.f;
#pragma unroll
        for (int v = 0; v < 8; ++v) s += rowmin[v];
        atomicAdd(acc, s * invCount);
    }
}

__global__ void init_kernel(float* acc) {
    if (threadIdx.x == 0) acc[0] = 0.0f;
}

__global__ __launch_bounds__(256)
void kp_kernel(const float* __restrict__ pred, const float* __restrict__ target,
               int n, float* __restrict__ acc)
{
    __shared__ float red[256];
    int i = threadIdx.x;
    float s = 0.f;
    for (int k = i; k < n; k += 256) { float d = pred[k] - target[k]; s += d * d; }
    red[i] = s;
    __syncthreads();
    for (int off = 128; off > 0; off >>= 1) {
        if (i < off) red[i] += red[i + off];
        __syncthreads();
    }
    if (i == 0) atomicAdd(acc, red[0] / (float)n);
}

__global__ void finalize_kernel(const float* __restrict__ acc, float* __restrict__ out) {
    if (threadIdx.x == 0) out[0] = acc[0];
}

extern "C" void kernel_launch(void* const* d_in, const int* in_sizes, int n_in,
                              void* d_out, int out_size, void* d_ws, size_t ws_size,
                              hipStream_t stream) {
    const float* pred   = (const float*)d_in[0];   // [8,10,3]
    const float* target = (const float*)d_in[1];   // [8,10,3]
    const float* coarse = (const float*)d_in[2];   // [8,3,1024]
    const float* fine   = (const float*)d_in[3];   // [8,3,4096]
    const float* gt     = (const float*)d_in[4];   // [8,4096,3]
    float* out = (float*)d_out;
    float* acc = (float*)d_ws;                     // single f32 accumulator

    const int B = 8, Nc = 1024, Nf = 4096, M = 4096;

    init_kernel<<<dim3(1), dim3(32), 0, stream>>>(acc);
    kp_kernel<<<dim3(1), dim3(256), 0, stream>>>(pred, target, in_sizes[0], acc);

    dim3 blk(128);  // 4 waves, 64 A-rows per block

    // coarse^T [B,Nc,3] strides (3*Nc, 1, Nc)  vs  gt [B,M,3] strides (3*M, 3, 1)
    chamfer_rowmin_kernel<<<dim3(Nc / 64, B), blk, 0, stream>>>(
        coarse, 3 * Nc, 1, Nc, Nc,   gt, 3 * M, 3, 1, M,   acc, 1.0f / (B * Nc));
    chamfer_rowmin_kernel<<<dim3(M / 64, B), blk, 0, stream>>>(
        gt, 3 * M, 3, 1, M,          coarse, 3 * Nc, 1, Nc, Nc,   acc, 1.0f / (B * M));

    // fine^T [B,Nf,3] strides (3*Nf, 1, Nf)  vs  gt
    chamfer_rowmin_kernel<<<dim3(Nf / 64, B), blk, 0, stream>>>(
        fine, 3 * Nf, 1, Nf, Nf,     gt, 3 * M, 3, 1, M,   acc, 1.0f / (B * Nf));
    chamfer_rowmin_kernel<<<dim3(M / 64, B), blk, 0, stream>>>(
        gt, 3 * M, 3, 1, M,          fine, 3 * Nf, 1, Nf, Nf,   acc, 1.0f / (B * M));

    finalize_kernel<<<dim3(1), dim3(32), 0, stream>>>(acc, out);
}